// CustomEmbeddingCollection_24412594111160
// MI455X (gfx1250) — compile-verified
//
#include <hip/hip_runtime.h>

// ---------------------------------------------------------------------------
// CDNA5 (gfx1250) embedding-cache lookup.
//   miss id x              -> row = cpu_weight[x]
//   hit slot s, tick<=nu   -> row = cpu_weight[unique_ids[tick[s]-1]]  (evicted+refilled)
//   hit slot s, otherwise  -> row = cache_data[s]
// unique_ids = ascending distinct miss ids (bitmap + popcount prefix scan,
// identical ranks to the reference's argsort-based uniquification).
// ---------------------------------------------------------------------------

#ifndef USE_ASYNC_LDS
#  if defined(__has_builtin)
#    if __has_builtin(__builtin_amdgcn_global_load_async_to_lds_b128) && \
        __has_builtin(__builtin_amdgcn_s_wait_asynccnt)
#      define USE_ASYNC_LDS 1
#    endif
#  endif
#endif
#ifndef USE_ASYNC_LDS
#  define USE_ASYNC_LDS 0
#endif

#define DIM      64   // floats per embedding row (fixed by reference)
#define CHUNKS   16   // float4 chunks per row
#define TILEROWS 16   // rows per 256-thread tile (16 rows * 16 chunks = 256)

// ---------------- K0: zero the miss-id bitmap ----------------
__global__ void k_init(unsigned* __restrict__ bitmap, int W) {
  int i = blockIdx.x * blockDim.x + threadIdx.x;
  if (i < W) bitmap[i] = 0u;
}

// ---------------- K1: mark distinct miss ids ----------------
__global__ void k_mark(const int* __restrict__ idx, const int* __restrict__ mapping,
                       unsigned* __restrict__ bitmap, int N) {
  int i = blockIdx.x * blockDim.x + threadIdx.x;
  if (i >= N) return;
  int id = idx[i];
  if (mapping[id] < 0)
    atomicOr(&bitmap[id >> 5], 1u << (id & 31));
}

// ---------------- K2: per-256-word popcount reduce ----------------
__global__ void k_reduce(const unsigned* __restrict__ bitmap,
                         int* __restrict__ blockSums, int W) {
  __shared__ int sh[256];
  int w = blockIdx.x * 256 + threadIdx.x;
  sh[threadIdx.x] = (w < W) ? __popc(bitmap[w]) : 0;
  __syncthreads();
  for (int o = 128; o > 0; o >>= 1) {
    if ((int)threadIdx.x < o) sh[threadIdx.x] += sh[threadIdx.x + o];
    __syncthreads();
  }
  if (threadIdx.x == 0) blockSums[blockIdx.x] = sh[0];
}

// ---------------- K3: single-block exclusive scan of block sums ----------------
__global__ void k_scan_sums(int* __restrict__ blockSums, int* __restrict__ numUniq, int SB) {
  __shared__ int sh[1024];
  int t = threadIdx.x;
  int v = (t < SB) ? blockSums[t] : 0;
  sh[t] = v;
  __syncthreads();
  for (int o = 1; o < 1024; o <<= 1) {
    int add = (t >= o) ? sh[t - o] : 0;
    __syncthreads();
    sh[t] += add;
    __syncthreads();
  }
  if (t < SB) blockSums[t] = sh[t] - v;   // exclusive block offset
  if (t == 1023) *numUniq = sh[1023];     // total distinct miss ids
}

// ---------------- K4: emit sorted distinct miss ids ----------------
__global__ void k_emit(const unsigned* __restrict__ bitmap, const int* __restrict__ blockOffs,
                       int* __restrict__ uniqIds, int W) {
  __shared__ int sh[256];
  int t = threadIdx.x;
  int w = blockIdx.x * 256 + t;
  unsigned bits = (w < W) ? bitmap[w] : 0u;
  int v = __popc(bits);
  sh[t] = v;
  __syncthreads();
  for (int o = 1; o < 256; o <<= 1) {
    int add = (t >= o) ? sh[t - o] : 0;
    __syncthreads();
    sh[t] += add;
    __syncthreads();
  }
  int r = blockOffs[blockIdx.x] + sh[t] - v;   // exclusive global rank
  int base = w << 5;
  while (bits) {
    int b = __ffs(bits) - 1;
    uniqIds[r++] = base + b;                   // ascending ids -> ascending ranks
    bits &= bits - 1u;
  }
}

// ---------------- K5: resolve source row per position + L2 prefetch ----------------
__global__ void k_resolve(const int* __restrict__ idx, const int* __restrict__ mapping,
                          const int* __restrict__ tick, const int* __restrict__ uniqIds,
                          const int* __restrict__ numUniq,
                          const float* __restrict__ cpu_w, const float* __restrict__ cache,
                          int* __restrict__ srcArr, int N) {
  int i = blockIdx.x * blockDim.x + threadIdx.x;
  if (i >= N) return;
  int id = idx[i];
  int s  = mapping[id];
  int nu = *numUniq;
  int src;
  if (s >= 0) {
    // evict[r] is the slot with access_tick == r+1 (ticks are a permutation).
    int pos = tick[s] - 1;
    src = (pos >= 0 && pos < nu) ? (-uniqIds[pos] - 1) : s;
  } else {
    src = -id - 1;                             // miss -> cpu_weight[id]
  }
  srcArr[i] = src;
  // Pre-stage the 256B row into L2 (192MB -> whole gathered set fits).
  const float* p = (src < 0) ? (cpu_w + (size_t)(-1 - src) * DIM)
                             : (cache + (size_t)src * DIM);
  __builtin_prefetch(p, 0, 1);        // global_prefetch_b8
  __builtin_prefetch(p + 32, 0, 1);   // second 128B line of the row
}

// ---------------- K6: gather rows to output ----------------
#if USE_ASYNC_LDS
typedef int v4i __attribute__((vector_size(16)));
typedef __attribute__((address_space(1))) v4i* gv4i_p;   // global int4*
typedef __attribute__((address_space(3))) v4i* lv4i_p;   // LDS int4*

__global__ void k_gather_async(const float* __restrict__ cpu_w, const float* __restrict__ cache,
                               const int* __restrict__ srcArr, float* __restrict__ out, int N) {
  __shared__ float4 lbuf[2][256];              // 2 x 16 rows x 256B (8KB, double buffered)
  const int tid = threadIdx.x;
  const int rit = tid >> 4;                    // row within tile
  const int ch  = tid & 15;                    // float4 chunk within row
  const int tiles   = (N + TILEROWS - 1) / TILEROWS;
  const int tstride = gridDim.x;
  int t = blockIdx.x;

  if (t < tiles) {                             // prologue: async-load tile t into buf0
    int row = min(t * TILEROWS + rit, N - 1);  // clamp keeps EXEC all-ones
    int s = srcArr[row];
    const float* p = (s < 0) ? (cpu_w + (size_t)(-1 - s) * DIM)
                             : (cache + (size_t)s * DIM);
    __builtin_amdgcn_global_load_async_to_lds_b128(
        (gv4i_p)(void*)(p + ch * 4), (lv4i_p)(void*)&lbuf[0][tid], 0, 0);
  }
  int buf = 0;
  while (t < tiles) {                          // uniform per block -> uniform ASYNCcnt
    int tn = t + tstride;
    if (tn < tiles) {                          // issue next tile, then drain to 1
      int row = min(tn * TILEROWS + rit, N - 1);
      int s = srcArr[row];
      const float* p = (s < 0) ? (cpu_w + (size_t)(-1 - s) * DIM)
                               : (cache + (size_t)s * DIM);
      __builtin_amdgcn_global_load_async_to_lds_b128(
          (gv4i_p)(void*)(p + ch * 4), (lv4i_p)(void*)&lbuf[buf ^ 1][tid], 0, 0);
      __builtin_amdgcn_s_wait_asynccnt(1);     // tile t is now in LDS
    } else {
      __builtin_amdgcn_s_wait_asynccnt(0);
    }
    __asm__ __volatile__("" ::: "memory");     // keep the ds read below the wait
    float4 v = lbuf[buf][tid];
    int row = t * TILEROWS + rit;
    if (row < N)
      ((float4*)out)[(size_t)row * CHUNKS + ch] = v;   // coalesced b128 stores
    t = tn;
    buf ^= 1;
  }
}
#endif

// Fallback: direct b128 gather (also used if async builtins are unavailable).
__global__ void k_gather(const float* __restrict__ cpu_w, const float* __restrict__ cache,
                         const int* __restrict__ srcArr, float* __restrict__ out, int N) {
  long long g = (long long)blockIdx.x * blockDim.x + threadIdx.x;
  int row = (int)(g >> 4);
  int ch  = (int)(g & 15);
  if (row >= N) return;
  int s = srcArr[row];
  const float4* p = (const float4*)((s < 0) ? (cpu_w + (size_t)(-1 - s) * DIM)
                                            : (cache + (size_t)s * DIM));
  ((float4*)out)[(size_t)row * CHUNKS + ch] = p[ch];
}

// ---------------------------------------------------------------------------
extern "C" void kernel_launch(void* const* d_in, const int* in_sizes, int n_in,
                              void* d_out, int out_size, void* d_ws, size_t ws_size,
                              hipStream_t stream) {
  (void)n_in; (void)out_size; (void)ws_size;
  const int*   indices = (const int*)  d_in[0];
  const float* cpu_w   = (const float*)d_in[1];
  const float* cache   = (const float*)d_in[2];
  const int*   mapping = (const int*)  d_in[3];
  const int*   tick    = (const int*)  d_in[4];
  // d_in[5] slot_to_id only affects mapping invalidation, which never reaches `out`.
  float* out = (float*)d_out;

  const int N  = in_sizes[0];            // 409600 flat lookups
  const int E  = in_sizes[3];            // vocab (mapping table length)
  const int W  = (E + 31) >> 5;          // bitmap words (62500)
  const int SB = (W + 255) >> 8;         // scan blocks (245, must be <= 1024)

  // workspace layout (ints), ~3.5MB total
  int* bitmap    = (int*)d_ws;
  int* blockSums = bitmap    + (((size_t)W + 63) & ~(size_t)63);
  int* numUniq   = blockSums + (((size_t)SB + 63) & ~(size_t)63);
  int* uniqIds   = numUniq   + 64;
  int* srcArr    = uniqIds   + (((size_t)N + 63) & ~(size_t)63);

  k_init     <<<(W + 255) / 256, 256, 0, stream>>>((unsigned*)bitmap, W);
  k_mark     <<<(N + 255) / 256, 256, 0, stream>>>(indices, mapping, (unsigned*)bitmap, N);
  k_reduce   <<<SB,              256, 0, stream>>>((const unsigned*)bitmap, blockSums, W);
  k_scan_sums<<<1,              1024, 0, stream>>>(blockSums, numUniq, SB);
  k_emit     <<<SB,              256, 0, stream>>>((const unsigned*)bitmap, blockSums, uniqIds, W);
  k_resolve  <<<(N + 255) / 256, 256, 0, stream>>>(indices, mapping, tick, uniqIds, numUniq,
                                                   cpu_w, cache, srcArr, N);
#if USE_ASYNC_LDS
  {
    int tiles   = (N + TILEROWS - 1) / TILEROWS;
    int gblocks = tiles < 8192 ? tiles : 8192;
    k_gather_async<<<gblocks, 256, 0, stream>>>(cpu_w, cache, srcArr, out, N);
  }
#else
  {
    long long chunksTot = (long long)N * CHUNKS;
    k_gather<<<(unsigned)((chunksTot + 255) / 256), 256, 0, stream>>>(cpu_w, cache, srcArr, out, N);
  }
#endif
}